// LoRA_FC_26714696581715
// MI455X (gfx1250) — compile-verified
//
#include <hip/hip_runtime.h>

typedef __attribute__((ext_vector_type(2))) float v2f;
typedef __attribute__((ext_vector_type(4))) float v4f;
typedef __attribute__((ext_vector_type(8))) float v8f;

#define DD 10000
#define RR 100

// Uniform K partition: 2500 quads total; every wave does KQ=312 quads (9984
// columns), remaining 4 quads (k = 9984..9999) handled by waves 0..3.
#define KQ 312
#define KTAIL_BASE 9984

// ---------------------------------------------------------------------------
// Kernel 1: t = A @ x   (100 x 10000) @ (10000) -> (100)
// Pure bandwidth (4 MB); float4 loads + LDS tree reduction per row.
// ---------------------------------------------------------------------------
__global__ __launch_bounds__(256) void lora_Ax(const float* __restrict__ A,
                                               const float* __restrict__ x,
                                               float* __restrict__ t) {
    __shared__ float red[256];
    const int  r    = blockIdx.x;
    const long base = (long)r * DD;

    float s = 0.f;
    for (int k4 = threadIdx.x; k4 < DD / 4; k4 += 256) {
        v4f a  = *(const v4f*)(A + base + 4 * k4);
        v4f xv = *(const v4f*)(x + 4 * k4);
        s += a.x * xv.x + a.y * xv.y + a.z * xv.z + a.w * xv.w;
    }
    red[threadIdx.x] = s;
    __syncthreads();
    for (int off = 128; off > 0; off >>= 1) {
        if (threadIdx.x < off) red[threadIdx.x] += red[threadIdx.x + off];
        __syncthreads();
    }
    if (threadIdx.x == 0) t[r] = red[0];
}

// ---------------------------------------------------------------------------
// Kernel 2: y = W @ x + B @ t   via V_WMMA_F32_16X16X4_F32.
// Block = 16 rows of y, 8 waves split K round-robin in quads of 4 columns.
// B-operand broadcasts x across all 16 columns -> every column of the C tile
// holds the GEMV result; we extract column N=0 (lanes 0 and 16).
// Constant trip count + dual accumulators -> unrolled, clause-batched loads
// with partial s_wait_loadcnt instead of a full stall per wmma.
// ---------------------------------------------------------------------------
__global__ __launch_bounds__(256) void lora_wmma_gemv(const float* __restrict__ W,
                                                      const float* __restrict__ Bm,
                                                      const float* __restrict__ x,
                                                      const float* __restrict__ t,
                                                      float* __restrict__ y) {
    __shared__ float red[8][16];

    const int lane = threadIdx.x & 31;
    const int wave = threadIdx.x >> 5;
    const int row0 = blockIdx.x << 4;

    const int m  = lane & 15;           // tile row (A operand)
    const int kh = (lane >> 4) << 1;    // K sub-offset: 0 (lanes 0-15) / 2 (16-31)

    const float* __restrict__ wrow = W + (long)(row0 + m) * DD + kh;

    v8f c0 = {0.f, 0.f, 0.f, 0.f, 0.f, 0.f, 0.f, 0.f};
    v8f c1 = {0.f, 0.f, 0.f, 0.f, 0.f, 0.f, 0.f, 0.f};

    // Wave `wave` owns K-quads {wave, wave+8, wave+16, ...}: 128-byte spans of
    // each row rotate across the 8 waves -> good line reuse in WGP$/L2.
    const float* __restrict__ wp = wrow + wave * 4;
    const float* __restrict__ xp = x + wave * 4 + kh;

#pragma unroll 4
    for (int i = 0; i < KQ; i += 2) {
        v2f a0 = *(const v2f*)(wp);
        v2f b0 = *(const v2f*)(xp);
        v2f a1 = *(const v2f*)(wp + 32);
        v2f b1 = *(const v2f*)(xp + 32);
        wp += 64;
        xp += 64;
        c0 = __builtin_amdgcn_wmma_f32_16x16x4_f32(false, a0, false, b0,
                                                   (short)0, c0, false, false);
        c1 = __builtin_amdgcn_wmma_f32_16x16x4_f32(false, a1, false, b1,
                                                   (short)0, c1, false, false);
    }

    // Tail: 4 remaining quads (columns 9984..9999) on waves 0..3 (wave-uniform
    // branch, EXEC stays all-ones inside).
    if (wave < 4) {
        const int k = KTAIL_BASE + wave * 4;
        v2f a = *(const v2f*)(wrow + k);
        v2f b = *(const v2f*)(x + k + kh);
        c0 = __builtin_amdgcn_wmma_f32_16x16x4_f32(false, a, false, b,
                                                   (short)0, c0, false, false);
    }

    // LoRA correction: rows of B (D x R) dotted with t (R) — wave 0 only.
    // R = 100 = 25 quads, constant trip count.
    if (wave == 0) {
        const float* __restrict__ brow = Bm + (long)(row0 + m) * RR + kh;
#pragma unroll 5
        for (int k = 0; k < RR; k += 4) {
            v2f a = *(const v2f*)(brow + k);
            v2f b = *(const v2f*)(t + k + kh);
            c1 = __builtin_amdgcn_wmma_f32_16x16x4_f32(false, a, false, b,
                                                       (short)0, c1, false, false);
        }
    }

    const v8f c = c0 + c1;

    // C/D layout: VGPR j, lanes 0-15 -> (M=j, N=lane); lanes 16-31 -> (M=j+8).
    // Column N==0 lives in lanes 0 (rows 0..7) and 16 (rows 8..15).
    if (m == 0) {
        const int rb = (lane >> 4) * 8;
#pragma unroll
        for (int j = 0; j < 8; ++j) red[wave][rb + j] = c[j];
    }
    __syncthreads();

    // Reduce the 8 per-wave partial tiles, store 16 outputs.
    if (threadIdx.x < 16) {
        float s = 0.f;
#pragma unroll
        for (int w = 0; w < 8; ++w) s += red[w][threadIdx.x];
        y[row0 + threadIdx.x] = s;
    }
}

// ---------------------------------------------------------------------------
// Host-side launcher. Input order from setup_inputs(): x, A, B, W (all f32).
// d_ws holds t (100 floats).
// ---------------------------------------------------------------------------
extern "C" void kernel_launch(void* const* d_in, const int* in_sizes, int n_in,
                              void* d_out, int out_size, void* d_ws, size_t ws_size,
                              hipStream_t stream) {
    (void)in_sizes; (void)n_in; (void)out_size; (void)ws_size;

    const float* x  = (const float*)d_in[0];
    const float* A  = (const float*)d_in[1];
    const float* Bm = (const float*)d_in[2];
    const float* W  = (const float*)d_in[3];
    float*       y  = (float*)d_out;
    float*       t  = (float*)d_ws;   // 100 floats of scratch

    lora_Ax<<<RR, 256, 0, stream>>>(A, x, t);
    lora_wmma_gemv<<<DD / 16, 256, 0, stream>>>(W, Bm, x, t, y);
}